// Memory_2259152797699
// MI455X (gfx1250) — compile-verified
//
#include <hip/hip_runtime.h>
#include <math.h>

// RWKV WKV scan, B=8, T=2048, H=2048, fp32.
// Chunked two-pass scan (16 chunks of 128 steps) for occupancy, with
// double-buffered GLOBAL_LOAD_ASYNC_TO_LDS_B128 prefetch of k/v tiles
// (gfx1250 async-copy path, tracked with ASYNCcnt / s_wait_asynccnt).
// Pass 3 maps blocks in reverse dispatch order so it consumes the tail of
// the k/v stream (still resident in the 192MB L2 after pass 1) first.

#define B_    8
#define T_    2048
#define H_    2048
#define NC    16
#define CHUNK (T_ / NC)     // 128
#define TS    16            // timesteps per LDS tile
#define NTILES (CHUNK / TS) // 8
#define BH    256           // threads per block == h-channels per block

// Copy one TS x BH fp32 tile (row stride H_ in global) into LDS using the
// gfx1250 async copy engine. 16B per lane per issue; 4 issues per thread.
__device__ __forceinline__ void async_copy_tile(const float* __restrict__ g,
                                                float* l, int tid) {
#pragma unroll
  for (int j = 0; j < (TS * BH) / (4 * BH); ++j) {
    const int u  = j * BH + tid;       // 16-byte unit index, lanes contiguous
    const int r  = u >> 6;             // 64 units per 256-float row
    const int c4 = (u & 63) << 2;      // float column
    const float* gp = g + (size_t)r * H_ + c4;
    const unsigned lp = (unsigned)(uintptr_t)(l + r * BH + c4); // LDS byte offset
    asm volatile("global_load_async_to_lds_b128 %0, %1, off"
                 :: "v"(lp), "v"(gp)
                 : "memory");
  }
}

// PASS3=false: compute per-chunk state summary (a,b,e) from zero state.
// PASS3=true : seed from prefix state, emit wkv outputs; blocks remapped in
//              reverse so early blocks hit the L2-resident tail of pass 1.
template <bool PASS3>
__global__ __launch_bounds__(256) void wkv_scan(
    const float* __restrict__ key, const float* __restrict__ val,
    const float* __restrict__ td, const float* __restrict__ tf,
    float* __restrict__ sa, float* __restrict__ sb, float* __restrict__ se,
    const float* __restrict__ pa, const float* __restrict__ pb,
    const float* __restrict__ pe, float* __restrict__ out) {
  __shared__ __align__(16) float kbuf[2][TS * BH];
  __shared__ __align__(16) float vbuf[2][TS * BH];

  const int tid = threadIdx.x;
  // Block -> data mapping (reversed for pass 3; scan order inside the chunk
  // is unchanged, so results are bit-identical to the forward mapping).
  const int xb = PASS3 ? (int)(gridDim.x - 1 - blockIdx.x) : (int)blockIdx.x;
  const int c  = PASS3 ? (NC - 1 - (int)blockIdx.y)        : (int)blockIdx.y;
  const int b  = PASS3 ? (B_ - 1 - (int)blockIdx.z)        : (int)blockIdx.z;

  const int h   = xb * BH + tid;
  const int t0  = c * CHUNK;
  const size_t base = ((size_t)b * T_ + t0) * H_ + (size_t)xb * BH;
  const size_t sIdx = ((size_t)b * NC + c) * H_ + h;

  const float u = tf[h];
  const float w = -__expf(td[h]);

  float a, bsum, e;
  if (PASS3) { a = pa[sIdx]; bsum = pb[sIdx]; e = pe[sIdx]; }
  else       { a = 0.f;      bsum = 0.f;      e = -INFINITY; }

  // Prologue: tile 0 in flight.
  async_copy_tile(key + base, kbuf[0], tid);
  async_copy_tile(val + base, vbuf[0], tid);

  size_t outIdx = base + tid;  // (b, t0, h)

  for (int tile = 0; tile < NTILES; ++tile) {
    const int cur = tile & 1;
    if (tile + 1 < NTILES) {
      // Issue next tile (8 async ops per wave), then wait so that only those
      // 8 remain outstanding -> current tile's 8 ops are complete (in-order).
      const size_t nb = base + (size_t)(tile + 1) * TS * H_;
      async_copy_tile(key + nb, kbuf[cur ^ 1], tid);
      async_copy_tile(val + nb, vbuf[cur ^ 1], tid);
      asm volatile("s_wait_asynccnt 0x8" ::: "memory");
    } else {
      asm volatile("s_wait_asynccnt 0x0" ::: "memory");
    }
    __syncthreads();  // all waves' LDS writes for current tile visible

#pragma unroll
    for (int r = 0; r < TS; ++r) {
      const float kt = kbuf[cur][r * BH + tid];
      const float vt = vbuf[cur][r * BH + tid];
      if (PASS3) {
        // output uses state from before this step (log-sum-exp stabilized)
        const float mo = fmaxf(e, u + kt);
        const float wt = __expf(u + kt - mo);
        const float sc = __expf(e - mo);
        out[outIdx] = (a * sc + wt * vt) / (bsum * sc + wt);
        outIdx += H_;
      }
      // state update with decay w
      const float m2 = fmaxf(e + w, kt);
      const float s1 = __expf(e + w - m2);
      const float s2 = __expf(kt - m2);
      a    = s1 * a + s2 * vt;
      bsum = s1 * bsum + s2;
      e    = m2;
    }
    __syncthreads();  // before next iteration overwrites buffer cur
  }

  if (!PASS3) { sa[sIdx] = a; sb[sIdx] = bsum; se[sIdx] = e; }
}

// Compose the NC chunk summaries per (b,h) into exclusive prefix states.
// Chunk map on the true state: A -> A*exp(CHUNK*w) + ca*exp(ce).
__global__ __launch_bounds__(256) void wkv_combine(
    const float* __restrict__ td,
    const float* __restrict__ sa, const float* __restrict__ sb,
    const float* __restrict__ se,
    float* __restrict__ pa, float* __restrict__ pb, float* __restrict__ pe) {
  const int idx = blockIdx.x * blockDim.x + threadIdx.x;  // over B*H
  const int b = idx / H_;
  const int h = idx - b * H_;
  const float w = -__expf(td[h]);
  const float d = (float)CHUNK * w;
  float a = 0.f, bsum = 0.f, e = -INFINITY;
#pragma unroll
  for (int c = 0; c < NC; ++c) {
    const size_t i = ((size_t)b * NC + c) * H_ + h;
    pa[i] = a; pb[i] = bsum; pe[i] = e;  // exclusive prefix
    const float ce = se[i];
    const float m  = fmaxf(e + d, ce);
    const float s1 = __expf(e + d - m);
    const float s2 = __expf(ce - m);
    a    = s1 * a + s2 * sa[i];
    bsum = s1 * bsum + s2 * sb[i];
    e    = m;
  }
}

extern "C" void kernel_launch(void* const* d_in, const int* in_sizes, int n_in,
                              void* d_out, int out_size, void* d_ws, size_t ws_size,
                              hipStream_t stream) {
  const float* key = (const float*)d_in[0];
  const float* val = (const float*)d_in[1];
  const float* td  = (const float*)d_in[2];
  const float* tf  = (const float*)d_in[3];
  float* out = (float*)d_out;

  const size_t S = (size_t)B_ * NC * H_;      // 262144 floats per array
  float* sa = (float*)d_ws;                   // chunk summaries
  float* sb = sa + S;
  float* se = sb + S;
  float* pa = se + S;                         // exclusive prefix states
  float* pb = pa + S;
  float* pe = pb + S;

  dim3 grid(H_ / BH, NC, B_);                 // 8 x 16 x 8 = 1024 blocks

  wkv_scan<false><<<grid, BH, 0, stream>>>(key, val, td, tf, sa, sb, se,
                                           nullptr, nullptr, nullptr, nullptr);
  wkv_combine<<<(B_ * H_) / 256, 256, 0, stream>>>(td, sa, sb, se, pa, pb, pe);
  wkv_scan<true><<<grid, BH, 0, stream>>>(key, val, td, tf,
                                          nullptr, nullptr, nullptr,
                                          pa, pb, pe, out);
}